// LearnedPrimalDual_16363825397989
// MI455X (gfx1250) — compile-verified
//
#include <hip/hip_runtime.h>
#include <hip/hip_bf16.h>
#include <math.h>

// ---------- CDNA5 WMMA types ----------
typedef __attribute__((ext_vector_type(2))) float v2f;
typedef __attribute__((ext_vector_type(8))) float v8f;

#define SDIM 512
#define ADIM 60
#define DDIM 729
#define NPIX (SDIM * SDIM)   // 262144
#define NSIN (ADIM * DDIM)   // 43740
#define LITER 10

// ---------------- utility kernels ----------------
__global__ void k_zero(float* p, int n) {
  int i = blockIdx.x * blockDim.x + threadIdx.x;
  if (i < n) p[i] = 0.f;
}

__global__ void k_trig(const float* __restrict__ theta, float* ca, float* sa) {
  int i = threadIdx.x;
  if (i < ADIM) { ca[i] = cosf(theta[i]); sa[i] = sinf(theta[i]); }
}

// Ram-Lak spatial kernel: rl[m] = irfft(2*rfftfreq(2048))[m], m in [0,728]
__global__ void k_ramp(float* rl) {
  int m = blockIdx.x * blockDim.x + threadIdx.x;
  if (m >= DDIM) return;
  float acc = 0.f;
  for (int j = 1; j < 1024; ++j) {
    int r = (j * m) & 2047;                       // exact phase mod n
    float ang = (float)r * (6.28318530717958647692f / 2048.f);
    acc += 2.f * ((float)j * (1.f / 1024.f)) * __cosf(ang);
  }
  acc += ((m & 1) ? -1.f : 1.f);                  // Nyquist bin, ramp=1.0
  rl[m] = acc * (1.f / 2048.f);
}

// ---------------- Radon forward: img[512,512] -> sino[60,729] ----------------
__global__ void k_radon_fwd(const float* __restrict__ img,
                            const float* __restrict__ ca, const float* __restrict__ sa,
                            float* __restrict__ out) {
  int i = blockIdx.x * blockDim.x + threadIdx.x;
  if (i >= NSIN) return;
  int a = i / DDIM, d = i - a * DDIM;
  const float c  = (SDIM - 1) * 0.5f;
  const float s  = (float)d - (DDIM - 1) * 0.5f;
  const float cc = ca[a], ss = sa[a];
  float sum = 0.f;
  for (int t = 0; t < SDIM; ++t) {
    float tt = (float)t - c;
    float x = s * cc - tt * ss + c;
    float y = s * ss + tt * cc + c;
    float xf = floorf(x), yf = floorf(y);
    int ix = (int)xf, iy = (int)yf;
    float wx = x - xf, wy = y - yf;
    bool x0ok = (ix >= 0)     && (ix < SDIM);
    bool x1ok = (ix + 1 >= 0) && (ix + 1 < SDIM);
    bool y0ok = (iy >= 0)     && (iy < SDIM);
    bool y1ok = (iy + 1 >= 0) && (iy + 1 < SDIM);
    float v00 = (x0ok && y0ok) ? img[iy * SDIM + ix]           : 0.f;
    float v01 = (x1ok && y0ok) ? img[iy * SDIM + ix + 1]       : 0.f;
    float v10 = (x0ok && y1ok) ? img[(iy + 1) * SDIM + ix]     : 0.f;
    float v11 = (x1ok && y1ok) ? img[(iy + 1) * SDIM + ix + 1] : 0.f;
    sum += (v00 * (1.f - wx) + v01 * wx) * (1.f - wy)
         + (v10 * (1.f - wx) + v11 * wx) * wy;
  }
  out[i] = sum;
}

// ---------------- sinogram filter (direct 729-tap correlation) ----------------
__global__ void k_filter(const float* __restrict__ h0, const float* __restrict__ rl,
                         float* __restrict__ out) {
  __shared__ float srl[DDIM];
  for (int i = threadIdx.x; i < DDIM; i += blockDim.x) srl[i] = rl[i];
  __syncthreads();
  int i = blockIdx.x * blockDim.x + threadIdx.x;
  if (i >= NSIN) return;
  int a = i / DDIM, d = i - a * DDIM;
  const float* row = h0 + a * DDIM;
  float sum = 0.f;
  for (int k = 0; k < DDIM; ++k) {
    int dd = d - k; if (dd < 0) dd = -dd;
    sum += row[k] * srl[dd];
  }
  out[i] = sum * (3.14159265358979323846f / (2.f * (float)ADIM));
}

// ---------------- backprojection: sino[60,729] -> img[512,512] ----------------
__global__ void k_radon_bp(const float* __restrict__ sino,
                           const float* __restrict__ ca, const float* __restrict__ sa,
                           float* __restrict__ out) {
  int i = blockIdx.x * blockDim.x + threadIdx.x;
  if (i >= NPIX) return;
  int y = i >> 9, x = i & (SDIM - 1);
  const float c  = (SDIM - 1) * 0.5f;
  const float dc = (DDIM - 1) * 0.5f;
  float X = (float)x - c, Y = (float)y - c;
  float sum = 0.f;
  for (int a = 0; a < ADIM; ++a) {
    float sidx = X * ca[a] + Y * sa[a] + dc;
    float pf = floorf(sidx);
    int p = (int)pf;
    float w = sidx - pf;
    const float* row = sino + a * DDIM;
    float v0 = (p >= 0     && p < DDIM)     ? row[p]     : 0.f;
    float v1 = (p + 1 >= 0 && p + 1 < DDIM) ? row[p + 1] : 0.f;
    sum += v0 * (1.f - w) + v1 * w;
  }
  out[i] = sum;
}

// ---------------- concat kernels ----------------
__global__ void k_concat_d(const float* __restrict__ h, const float* __restrict__ opf,
                           const float* __restrict__ g, float* __restrict__ cat) {
  int i = blockIdx.x * blockDim.x + threadIdx.x;
  if (i >= 7 * NSIN) return;
  int c = i / NSIN, r = i - c * NSIN;
  cat[i] = (c < 5) ? h[i] : ((c == 5) ? opf[r] : g[r]);
}

__global__ void k_concat_p(const float* __restrict__ f, const float* __restrict__ opadj,
                           float* __restrict__ cat) {
  int i = blockIdx.x * blockDim.x + threadIdx.x;
  if (i >= 6 * NPIX) return;
  cat[i] = (i < 5 * NPIX) ? f[i] : opadj[i - 5 * NPIX];
}

__global__ void k_out3(const float* __restrict__ f0, float* __restrict__ out) {
  int i = blockIdx.x * blockDim.x + threadIdx.x;
  if (i >= 3 * NPIX) return;
  out[i] = f0[i & (NPIX - 1)];
}

// ---------------- 3x3 SAME conv as implicit GEMM via V_WMMA_F32_16X16X4_F32 ----
// D(16 couts x 16 pixels) += A(16 couts x 4 k) * B(4 k x 16 pixels), K = Cin*9.
// ISA fp32 fragment layouts (CDNA5 ISA 7.12.2):
//   A 16x4 : lane = M + 16*(K>>1), vgpr = K&1
//   B 4x16 : lane = N + 16*(K>>1), vgpr = K&1
//   C/D    : vgpr r -> M = r + 8*(lane>>4), N = lane&15
// Block = 128 threads = 4 waves; block covers 64 consecutive pixels of one row.
// LDS holds (a) zero-padded im2col patch CIN x 3 x 66, (b) weights in k-pair-
// major layout so each A fragment is a single aligned ds_load_b64 at a
// constant immediate offset. Fully unrolled K loop (compile-time CIN).

__host__ __device__ constexpr int boffc(int k, int K9) {
  // LDS patch offset for im2col slot k (0 if padded slot; its A rows are 0)
  return (k >= K9) ? 0
                   : ((k / 9) * (3 * 66) + (((k % 9) / 3) * 66) + ((k % 9) % 3));
}

template <int CIN, int COUT>
__global__ void __launch_bounds__(128)
k_conv3x3_wmma(const float* __restrict__ in, const float* __restrict__ w,
               const float* __restrict__ bias, const float* __restrict__ alphaPtr,
               float* __restrict__ out, int H, int W, int accumulate) {
  constexpr int K9  = CIN * 9;
  constexpr int K9P = (K9 + 3) & ~3;
  constexpr int PW  = 66;                        // 64 pixels + 2 halo
  constexpr int NCT = (COUT + 15) / 16;          // 1 or 2 cout tiles

  __shared__ float patch[CIN * 3 * PW];
  __shared__ float swT[K9P * 32];                // [(k>>1)][m][k&1]

  const int tid    = threadIdx.x;
  const int xTiles = (W + 63) >> 6;
  const int y      = blockIdx.x / xTiles;
  const int xbase  = (blockIdx.x - y * xTiles) << 6;

  // ---- stage weights (transpose to k-major pairs, zero-pad K and M) ----
  for (int i = tid; i < K9P * 32; i += 128) {
    int k = i >> 5, m = i & 31;
    float v = (k < K9 && m < COUT) ? w[m * K9 + k] : 0.f;
    swT[(k >> 1) * 64 + m * 2 + (k & 1)] = v;
  }
  // ---- stage input patch with SAME zero padding resolved here ----
  for (int i = tid; i < CIN * 3 * PW; i += 128) {
    int cin = i / (3 * PW);
    int r   = i - cin * 3 * PW;
    int ky  = r / PW;
    int col = r - ky * PW;
    int iy = y + ky - 1;
    int ix = xbase + col - 1;
    patch[i] = (iy >= 0 && iy < H && ix >= 0 && ix < W)
                 ? in[cin * H * W + iy * W + ix] : 0.f;
  }
  __syncthreads();

  const int lane = tid & 31;
  const int wv   = tid >> 5;
  const int hl   = lane >> 4;
  const int l15  = lane & 15;
  const int pb   = wv * 16 + l15;                // lane's patch column
  const float* aptr = swT + hl * 64 + l15 * 2;   // A fragment base (pairs)
  const float* pptr = patch + pb;

  v8f acc0 = {};
  v8f acc1 = {};

#pragma unroll
  for (int ks = 0; ks < K9P; ks += 4) {
    // half-wave k selection: lanes 0-15 take k=ks,ks+1; lanes 16-31 take ks+2,ks+3
    const int o0 = hl ? boffc(ks + 2, K9) : boffc(ks + 0, K9);
    const int o1 = hl ? boffc(ks + 3, K9) : boffc(ks + 1, K9);
    v2f B;
    B.x = pptr[o0];
    B.y = pptr[o1];
    v2f A0 = *(const v2f*)(aptr + ks * 32);          // (k0,m),(k1,m)
    acc0 = __builtin_amdgcn_wmma_f32_16x16x4_f32(false, A0, false, B,
                                                 (short)0, acc0, false, false);
    if (NCT > 1) {
      v2f A1 = *(const v2f*)(aptr + ks * 32 + 32);   // rows m+16
      acc1 = __builtin_amdgcn_wmma_f32_16x16x4_f32(false, A1, false, B,
                                                   (short)0, acc1, false, false);
    }
  }

  // ---- epilogue: bias + PReLU + (optional) residual accumulate ----
  const int xo = xbase + wv * 16 + l15;
  const float alpha = alphaPtr ? alphaPtr[0] : 0.f;
  if (xo < W) {
#pragma unroll
    for (int r = 0; r < 8; ++r) {
      const int m = r + 8 * hl;
      if (m < COUT) {
        float val = acc0[r] + bias[m];
        if (alphaPtr) val = (val >= 0.f) ? val : alpha * val;
        float* dst = out + m * H * W + y * W + xo;
        if (accumulate) *dst += val; else *dst = val;
      }
      if (NCT > 1) {
        const int m2 = 16 + m;
        if (m2 < COUT) {
          float val = acc1[r] + bias[m2];
          if (alphaPtr) val = (val >= 0.f) ? val : alpha * val;
          float* dst = out + m2 * H * W + y * W + xo;
          if (accumulate) *dst += val; else *dst = val;
        }
      }
    }
  }
}

// ---------------- host-side orchestration ----------------
static inline int cdiv(int a, int b) { return (a + b - 1) / b; }

static void launch_conv(const float* in, const float* w, const float* b,
                        const float* alpha, float* out, int Cin, int Cout,
                        int H, int W, int accumulate, hipStream_t stream) {
  int blocks = H * cdiv(W, 64);
  if (Cin == 7 && Cout == 32)
    k_conv3x3_wmma<7, 32><<<blocks, 128, 0, stream>>>(in, w, b, alpha, out, H, W, accumulate);
  else if (Cin == 32 && Cout == 32)
    k_conv3x3_wmma<32, 32><<<blocks, 128, 0, stream>>>(in, w, b, alpha, out, H, W, accumulate);
  else if (Cin == 32 && Cout == 5)
    k_conv3x3_wmma<32, 5><<<blocks, 128, 0, stream>>>(in, w, b, alpha, out, H, W, accumulate);
  else if (Cin == 6 && Cout == 32)
    k_conv3x3_wmma<6, 32><<<blocks, 128, 0, stream>>>(in, w, b, alpha, out, H, W, accumulate);
}

extern "C" void kernel_launch(void* const* d_in, const int* in_sizes, int n_in,
                              void* d_out, int out_size, void* d_ws, size_t ws_size,
                              hipStream_t stream) {
  (void)in_sizes; (void)n_in; (void)out_size; (void)ws_size;

  const float* g     = (const float*)d_in[2];
  const float* theta = (const float*)d_in[3];
  const float* dw1 = (const float*)d_in[5];
  const float* db1 = (const float*)d_in[6];
  const float* da1 = (const float*)d_in[7];
  const float* dw2 = (const float*)d_in[8];
  const float* db2 = (const float*)d_in[9];
  const float* da2 = (const float*)d_in[10];
  const float* dw3 = (const float*)d_in[11];
  const float* db3 = (const float*)d_in[12];
  const float* pw1 = (const float*)d_in[13];
  const float* pb1 = (const float*)d_in[14];
  const float* pa1 = (const float*)d_in[15];
  const float* pw2 = (const float*)d_in[16];
  const float* pb2 = (const float*)d_in[17];
  const float* pa2 = (const float*)d_in[18];
  const float* pw3 = (const float*)d_in[19];
  const float* pb3 = (const float*)d_in[20];

  // scratch layout (floats), ~82 MB total
  float* ws    = (float*)d_ws;
  float* f     = ws;                 // 5 * NPIX
  float* h     = f     + 5 * NPIX;   // 5 * NSIN
  float* cat   = h     + 5 * NSIN;   // 7 * NPIX (covers 6*NPIX primal / 7*NSIN dual)
  float* tA    = cat   + 7 * NPIX;   // 32 * NPIX
  float* tB    = tA    + 32 * NPIX;  // 32 * NPIX
  float* OpF   = tB    + 32 * NPIX;  // NSIN
  float* h1    = OpF   + NSIN;       // NSIN
  float* OpAdj = h1    + NSIN;       // NPIX
  float* rl    = OpAdj + NPIX;       // 1024
  float* ca    = rl    + 1024;       // 64
  float* sa    = ca    + 64;         // 64

  k_zero<<<cdiv(5 * NPIX, 256), 256, 0, stream>>>(f, 5 * NPIX);
  k_zero<<<cdiv(5 * NSIN, 256), 256, 0, stream>>>(h, 5 * NSIN);
  k_trig<<<1, 64, 0, stream>>>(theta, ca, sa);
  k_ramp<<<cdiv(DDIM, 128), 128, 0, stream>>>(rl);

  for (int i = 0; i < LITER; ++i) {
    // ---- dual step ----
    k_radon_fwd<<<cdiv(NSIN, 128), 128, 0, stream>>>(f + NPIX, ca, sa, OpF);
    k_concat_d<<<cdiv(7 * NSIN, 256), 256, 0, stream>>>(h, OpF, g, cat);
    launch_conv(cat, dw1 + i * 32 * 7 * 9, db1 + i * 32, da1 + i, tA, 7, 32, ADIM, DDIM, 0, stream);
    launch_conv(tA,  dw2 + i * 32 * 32 * 9, db2 + i * 32, da2 + i, tB, 32, 32, ADIM, DDIM, 0, stream);
    launch_conv(tB,  dw3 + i * 5 * 32 * 9,  db3 + i * 5,  nullptr, h,  32, 5,  ADIM, DDIM, 1, stream);

    // ---- filter + backprojection ----
    k_filter<<<cdiv(NSIN, 256), 256, 0, stream>>>(h, rl, h1);
    k_radon_bp<<<cdiv(NPIX, 256), 256, 0, stream>>>(h1, ca, sa, OpAdj);

    // ---- primal step ----
    k_concat_p<<<cdiv(6 * NPIX, 256), 256, 0, stream>>>(f, OpAdj, cat);
    launch_conv(cat, pw1 + i * 32 * 6 * 9,  pb1 + i * 32, pa1 + i, tA, 6, 32, SDIM, SDIM, 0, stream);
    launch_conv(tA,  pw2 + i * 32 * 32 * 9, pb2 + i * 32, pa2 + i, tB, 32, 32, SDIM, SDIM, 0, stream);
    launch_conv(tB,  pw3 + i * 5 * 32 * 9,  pb3 + i * 5,  nullptr, f,  32, 5,  SDIM, SDIM, 1, stream);
  }

  k_out3<<<cdiv(3 * NPIX, 256), 256, 0, stream>>>(f, (float*)d_out);
}